// BaseRGCNPairModel_558345748776
// MI455X (gfx1250) — compile-verified
//
#include <hip/hip_runtime.h>
#include <hip/hip_bf16.h>

// MI455X (gfx1250) implementation of 2-layer RGCN + pair scorer.
// Matrix ops: native f32 WMMA (v_wmma_f32_16x16x4_f32, wave32).
// A-tile staging: async global->LDS (ASYNCcnt) via
// __builtin_amdgcn_global_load_async_to_lds_b128.

typedef __attribute__((ext_vector_type(2))) float v2f;
typedef __attribute__((ext_vector_type(8))) float v8f;
typedef int v4i_128 __attribute__((vector_size(16)));

#define RGCN_N 100000
#define RGCN_R 8
#define RGCN_D 128
#define RGCN_E 600000
#define RGCN_P 100000

#if defined(__has_builtin)
#if __has_builtin(__builtin_amdgcn_global_load_async_to_lds_b128)
#define HAS_ASYNC_LDS 1
#endif
#if __has_builtin(__builtin_amdgcn_s_wait_asynccnt)
#define HAS_WAIT_ASYNC 1
#endif
#endif

__device__ __forceinline__ void wait_asynccnt0() {
#if defined(HAS_WAIT_ASYNC)
  __builtin_amdgcn_s_wait_asynccnt(0);
#else
  asm volatile("s_wait_asynccnt 0x0" ::: "memory");
#endif
}

// ---------------------------------------------------------------------------
// Utility kernels
// ---------------------------------------------------------------------------
__global__ void fill_zero_kernel(float* __restrict__ p, size_t n) {
  size_t i = (size_t)blockIdx.x * 256u + threadIdx.x;
  if (i < n) p[i] = 0.0f;
}

__global__ void relu_kernel(float* __restrict__ p, size_t n) {
  size_t i = (size_t)blockIdx.x * 256u + threadIdx.x;
  if (i < n) p[i] = fmaxf(p[i], 0.0f);
}

// Per-relation in-degree: deg[r*N + dst] += 1
__global__ void count_deg_kernel(const int* __restrict__ ei,
                                 const int* __restrict__ et,
                                 float* __restrict__ deg, int E, int N) {
  int e = blockIdx.x * 256 + threadIdx.x;
  if (e >= E) return;
  int dst = ei[E + e];
  int r = et[e];
  atomicAdd(&deg[(size_t)r * N + dst], 1.0f);
}

// ---------------------------------------------------------------------------
// Relation transform GEMM: for s in [0,R): xw[s] = x @ rw[s]
//                          s == R:        base   = x @ sw + bias
// grid = (N/16), block = 256 (8 waves, each owns one 16-wide N tile).
// The 16x128 A tile is staged to LDS ONCE and reused for all 9 weights.
// ---------------------------------------------------------------------------
__global__ __launch_bounds__(256) void rgcn_transform_kernel(
    const float* __restrict__ x,    // [N, D]
    const float* __restrict__ rw,   // [R, D, D]
    const float* __restrict__ sw,   // [D, D]
    const float* __restrict__ bias, // [D]
    float* __restrict__ xw,         // [R, N, D]
    float* __restrict__ base)       // [N, D]
{
  constexpr int D = RGCN_D;
  constexpr int LDA = 132;          // pad: 132 % 64 == 4 -> no bank conflicts
  __shared__ float As[16 * LDA];

  const int mt = blockIdx.x;        // M tile (16 rows)
  const int tid = threadIdx.x;

  // Prefetch first weight matrix (64KB: 256 threads x 256B)
  __builtin_prefetch(rw + tid * 64, 0, 0);

  // Stage A tile 16x128 into LDS (each thread moves 2 x 16B)
  {
    const int idx0 = tid * 2;
#pragma unroll
    for (int i = 0; i < 2; ++i) {
      int idx = idx0 + i;           // 0..511
      int row = idx >> 5;           // 32 float4 per row
      int c4 = idx & 31;
      const float* gsrc = &x[((size_t)(mt * 16 + row)) * D + c4 * 4];
      float* ldst = &As[row * LDA + c4 * 4];
#if defined(HAS_ASYNC_LDS)
      __builtin_amdgcn_global_load_async_to_lds_b128(
          (v4i_128*)gsrc, (v4i_128*)ldst, 0, 0);
#else
      *(float4*)ldst = *(const float4*)gsrc;
#endif
    }
#if defined(HAS_ASYNC_LDS)
    wait_asynccnt0();
#endif
  }
  __syncthreads();

  const int wv = tid >> 5;          // wave -> N tile
  const int lane = tid & 31;
  const int m = lane & 15;          // A row / B column-within-tile
  const int koff = (lane >> 4) << 1; // 0 or 2 (upper half-wave -> K+2)
  const int n0 = wv * 16;
  const int nc = n0 + m;            // B column
  const int row0 = (lane >> 4) << 3;
  const int col = n0 + (lane & 15);

#pragma unroll 1
  for (int s = 0; s <= RGCN_R; ++s) {
    const float* __restrict__ W =
        (s < RGCN_R) ? (rw + (size_t)s * D * D) : sw;

    v8f acc = {};
#pragma unroll
    for (int k0 = 0; k0 < D; k0 += 4) {
      // A 16x4 f32 frag: lane m holds K = k0+koff, k0+koff+1
      v2f a = *(const v2f*)&As[m * LDA + k0 + koff];
      // B 4x16 f32 frag: lane holds rows K = k0+koff, k0+koff+1 at col nc
      v2f b;
      b[0] = W[(size_t)(k0 + koff) * D + nc];
      b[1] = W[(size_t)(k0 + koff + 1) * D + nc];
      acc = __builtin_amdgcn_wmma_f32_16x16x4_f32(
          false, a, false, b, (short)0, acc, false, false);
    }

    // C/D layout: lane l, vgpr v -> row = v + 8*(l>=16), col = l & 15
    if (s < RGCN_R) {
      float* __restrict__ o = xw + ((size_t)s * RGCN_N + (size_t)mt * 16) * D;
#pragma unroll
      for (int v = 0; v < 8; ++v) {
        o[(size_t)(row0 + v) * D + col] = acc[v];
      }
    } else {
      const float bv = bias[col];
      float* __restrict__ o = base + (size_t)mt * 16 * D;
#pragma unroll
      for (int v = 0; v < 8; ++v) {
        o[(size_t)(row0 + v) * D + col] = acc[v] + bv;
      }
    }
  }
}

// ---------------------------------------------------------------------------
// Edge scatter: out[dst] += xw[etype][src] / max(deg[etype,dst], 1)
// one wave32 per edge, 4 floats per lane
// ---------------------------------------------------------------------------
__global__ __launch_bounds__(256) void scatter_edges_kernel(
    const int* __restrict__ ei,   // [2, E]
    const int* __restrict__ et,   // [E]
    const float* __restrict__ xw, // [R, N, D]
    const float* __restrict__ deg,// [R, N]
    float* __restrict__ out,      // [N, D]
    int E, int N)
{
  int gid = blockIdx.x * 256 + threadIdx.x;
  int e = gid >> 5;
  int lane = threadIdx.x & 31;
  if (e >= E) return;
  int src = ei[e];
  int dst = ei[E + e];
  int r = et[e];
  float inv = 1.0f / fmaxf(deg[(size_t)r * N + dst], 1.0f);
  const float4 v =
      *(const float4*)&xw[(((size_t)r * N + src) << 7) + lane * 4];
  float* __restrict__ o = out + ((size_t)dst << 7) + lane * 4;
  atomicAdd(o + 0, v.x * inv);
  atomicAdd(o + 1, v.y * inv);
  atomicAdd(o + 2, v.z * inv);
  atomicAdd(o + 3, v.w * inv);
}

// ---------------------------------------------------------------------------
// Pair scorer: logits = relu([dh|sh|dh*sh] @ pw1 + pb1) @ pw2 + pb2
// 16 pairs per block; feat tile staged in LDS; 8 waves WMMA over K=384;
// fused ReLU + pw2 dot via 16-lane shuffle reduction + LDS atomics.
// ---------------------------------------------------------------------------
__global__ __launch_bounds__(256) void pair_score_kernel(
    const float* __restrict__ h,   // [N, 128]
    const int* __restrict__ drug,  // [P]
    const int* __restrict__ dis,   // [P]
    const float* __restrict__ pw1, // [384, 128]
    const float* __restrict__ pb1, // [128]
    const float* __restrict__ pw2, // [128]
    const float* __restrict__ pb2, // [1]
    float* __restrict__ out, int P)
{
  constexpr int D = RGCN_D;
  constexpr int K = 3 * RGCN_D;    // 384
  constexpr int LDF = 388;         // 388 % 64 == 4 -> conflict-free
  __shared__ float F[16 * LDF];
  __shared__ float partial[16];
  __shared__ int dIdx[16], sIdx[16];

  const int p0 = blockIdx.x * 16;
  const int tid = threadIdx.x;

  if (tid < 16) {
    int p = min(p0 + tid, P - 1);
    dIdx[tid] = drug[p];
    sIdx[tid] = dis[p];
    partial[tid] = pb2[0];
  }
  __syncthreads();

  // Stage feat tile: 16 pairs x 384 (dh | sh | dh*sh)
  for (int i = tid; i < 16 * D; i += 256) {
    int pr = i >> 7;   // pair row
    int c = i & 127;   // feature col within section
    float dv = h[(size_t)dIdx[pr] * D + c];
    float sv = h[(size_t)sIdx[pr] * D + c];
    F[pr * LDF + c] = dv;
    F[pr * LDF + 128 + c] = sv;
    F[pr * LDF + 256 + c] = dv * sv;
  }
  __syncthreads();

  const int wv = tid >> 5;
  const int lane = tid & 31;
  const int m = lane & 15;
  const int koff = (lane >> 4) << 1;
  const int n0 = wv * 16;
  const int nc = n0 + m;

  v8f acc = {};
#pragma unroll 8
  for (int k0 = 0; k0 < K; k0 += 4) {
    v2f a = *(const v2f*)&F[m * LDF + k0 + koff];
    v2f b;
    b[0] = pw1[(size_t)(k0 + koff) * D + nc];
    b[1] = pw1[(size_t)(k0 + koff + 1) * D + nc];
    acc = __builtin_amdgcn_wmma_f32_16x16x4_f32(
        false, a, false, b, (short)0, acc, false, false);
  }

  const float w2 = pw2[n0 + (lane & 15)];
  const float b1v = pb1[n0 + (lane & 15)];
#pragma unroll
  for (int v = 0; v < 8; ++v) {
    float hv = fmaxf(acc[v] + b1v, 0.0f);
    float t = hv * w2;
    // reduce across the 16 lanes that hold one row's columns
    t += __shfl_xor(t, 1, 32);
    t += __shfl_xor(t, 2, 32);
    t += __shfl_xor(t, 4, 32);
    t += __shfl_xor(t, 8, 32);
    if ((lane & 15) == 0) {
      int row = v + ((lane >> 4) << 3);
      atomicAdd(&partial[row], t);
    }
  }
  __syncthreads();
  if (tid < 16 && (p0 + tid) < P) out[p0 + tid] = partial[tid];
}

// ---------------------------------------------------------------------------
// Host-side launch
// ---------------------------------------------------------------------------
extern "C" void kernel_launch(void* const* d_in, const int* in_sizes, int n_in,
                              void* d_out, int out_size, void* d_ws,
                              size_t ws_size, hipStream_t stream) {
  (void)in_sizes; (void)n_in; (void)out_size; (void)ws_size;
  const int* edge_index = (const int*)d_in[0];    // [2, E]
  const int* edge_type = (const int*)d_in[1];     // [E]
  const int* drug_index = (const int*)d_in[2];    // [P]
  const int* disease_index = (const int*)d_in[3]; // [P]
  const float* emb = (const float*)d_in[4];       // [N, D]
  const float* rw1 = (const float*)d_in[5];       // [R, D, D]
  const float* sl1 = (const float*)d_in[6];       // [D, D]
  const float* b1 = (const float*)d_in[7];        // [D]
  const float* rw2 = (const float*)d_in[8];
  const float* sl2 = (const float*)d_in[9];
  const float* b2 = (const float*)d_in[10];
  const float* pw1 = (const float*)d_in[11];      // [384, 128]
  const float* pb1 = (const float*)d_in[12];      // [128]
  const float* pw2 = (const float*)d_in[13];      // [128]
  const float* pb2 = (const float*)d_in[14];      // [1]

  constexpr int N = RGCN_N, R = RGCN_R, D = RGCN_D, E = RGCN_E, P = RGCN_P;

  // Workspace layout
  float* xw = (float*)d_ws;                       // R*N*D   (409.6 MB)
  float* h1 = xw + (size_t)R * N * D;             // N*D
  float* h2 = h1 + (size_t)N * D;                 // N*D
  float* deg = h2 + (size_t)N * D;                // R*N

  // Degrees (graph is static: compute once, reuse in both layers)
  {
    size_t n = (size_t)R * N;
    fill_zero_kernel<<<(unsigned)((n + 255) / 256), 256, 0, stream>>>(deg, n);
    count_deg_kernel<<<(E + 255) / 256, 256, 0, stream>>>(edge_index,
                                                          edge_type, deg, E, N);
  }

  int scatter_blocks = (E * 32 + 255) / 256;
  size_t nd = (size_t)N * D;

  // ---- Layer 1 ----
  rgcn_transform_kernel<<<N / 16, 256, 0, stream>>>(emb, rw1, sl1, b1, xw, h1);
  scatter_edges_kernel<<<scatter_blocks, 256, 0, stream>>>(
      edge_index, edge_type, xw, deg, h1, E, N);
  relu_kernel<<<(unsigned)((nd + 255) / 256), 256, 0, stream>>>(h1, nd);

  // ---- Layer 2 ----
  rgcn_transform_kernel<<<N / 16, 256, 0, stream>>>(h1, rw2, sl2, b2, xw, h2);
  scatter_edges_kernel<<<scatter_blocks, 256, 0, stream>>>(
      edge_index, edge_type, xw, deg, h2, E, N);

  // ---- Pair scorer ----
  pair_score_kernel<<<(P + 15) / 16, 256, 0, stream>>>(
      h2, drug_index, disease_index, pw1, pb1, pw2, pb2, (float*)d_out, P);
}